// QuantizerModule_42210938585803
// MI455X (gfx1250) — compile-verified
//
#include <hip/hip_runtime.h>

#define N_X   262144
#define E_DIM 128
#define N_E   1024

typedef __attribute__((ext_vector_type(2))) float v2f;
typedef __attribute__((ext_vector_type(8))) float v8f;
typedef __attribute__((ext_vector_type(4))) int   v4i;

#define AS1 __attribute__((address_space(1)))
#define AS3 __attribute__((address_space(3)))

#if __has_builtin(__builtin_amdgcn_global_load_async_to_lds_b128)
#define VQ_ASYNC_LDS 1
#else
#define VQ_ASYNC_LDS 0
#endif

// ---------------------------------------------------------------------------
// Kernel 1: wnorm[k] = ||W_k||^2   (one wave per codebook row, 4 waves/block)
// ---------------------------------------------------------------------------
__global__ __launch_bounds__(128) void vq_wnorm_kernel(
    const float* __restrict__ W, float* __restrict__ wnorm) {
  const int wave = threadIdx.x >> 5;
  const int lane = threadIdx.x & 31;
  const int row  = blockIdx.x * 4 + wave;
  const float4* wr = (const float4*)(W + (long)row * E_DIM);
  float4 v = wr[lane];                      // 32 lanes * float4 = 128 floats
  float s = v.x * v.x + v.y * v.y + v.z * v.z + v.w * v.w;
#pragma unroll
  for (int off = 16; off > 0; off >>= 1) s += __shfl_xor(s, off, 32);
  if (lane == 0) wnorm[row] = s;
}

// ---------------------------------------------------------------------------
// Async (or fallback sync) 8KB W-tile copy into LDS: 128 threads x 4 x b128.
// ---------------------------------------------------------------------------
__device__ __forceinline__ void vq_tile_copy(const float* __restrict__ gsrc,
                                             float* lds_dst, int tid) {
#if VQ_ASYNC_LDS
#pragma unroll
  for (int i = 0; i < 4; ++i) {
    const float* g = gsrc + (size_t)(tid + i * 128) * 4;
    float*       l = lds_dst + (size_t)(tid + i * 128) * 4;
    __builtin_amdgcn_global_load_async_to_lds_b128(
        (AS1 v4i*)(unsigned long long)(__SIZE_TYPE__)g,
        (AS3 v4i*)(unsigned)(__SIZE_TYPE__)l, 0, 0);
  }
#else
  const float4* src = (const float4*)gsrc;
  float4*       dst = (float4*)lds_dst;
#pragma unroll
  for (int i = 0; i < 4; ++i) dst[tid + i * 128] = src[tid + i * 128];
#endif
}

__device__ __forceinline__ void vq_wait_async() {
#if VQ_ASYNC_LDS
#if __has_builtin(__builtin_amdgcn_s_wait_asynccnt)
  __builtin_amdgcn_s_wait_asynccnt(0);
#else
  asm volatile("s_wait_asynccnt 0x0" ::: "memory");
#endif
#endif
}

// ---------------------------------------------------------------------------
// Kernel 2: fused distance + argmin via V_WMMA_F32_16X16X4_F32.
//   Block = 128 threads = 4 waves; each wave owns 16 x rows (block owns 64).
//   Double-buffered LDS W tiles filled by async global->LDS DMA; tile t+1's
//   DMA overlaps tile t's 32-WMMA chain. Argmin stays per-lane because the
//   D tile is [codes(M) x xcols(N)] and lanes L / L^16 share an x column.
// ---------------------------------------------------------------------------
__global__ __launch_bounds__(128) void vq_argmin_kernel(
    const float* __restrict__ x, const float* __restrict__ W,
    const float* __restrict__ wnorm, int* __restrict__ idx_out) {
  __shared__ float wtile[2][16 * E_DIM];    // 2 x 8 KB of the 320 KB WGP LDS

  const int  tid  = threadIdx.x;
  const int  wave = tid >> 5;
  const int  lane = tid & 31;
  const int  half = lane >> 4;              // K-half / M-half selector
  const int  col  = lane & 15;              // x column (N) and A row (M)
  const long rowBase = (long)blockIdx.x * 64 + (long)wave * 16;

  // Preload all B fragments (x) for this wave's 16 columns: 32 K-steps,
  // 2 fp32 each -> 64 VGPRs. Lane holds column N=col; VGPR v holds
  // K = half*2 + v within each 4-wide K slab.
  v2f bfrag[32];
  {
    const float* xp = x + (rowBase + col) * E_DIM + half * 2;
#pragma unroll
    for (int kk = 0; kk < 32; ++kk) bfrag[kk] = *(const v2f*)(xp + kk * 4);
  }

  // Kick off DMA for tile 0.
  vq_tile_copy(W, &wtile[0][0], tid);

  float minv = 3.402823466e38f;
  int   mini = 0;

  for (int t = 0; t < N_E / 16; ++t) {
    vq_wait_async();      // this wave's portion of tile t is in LDS
    __syncthreads();      // all waves' portions landed; prev tile fully read

    if (t + 1 < N_E / 16) // overlap tile t+1 DMA with tile t compute
      vq_tile_copy(W + (long)(t + 1) * 16 * E_DIM, &wtile[(t + 1) & 1][0], tid);

    // A layout: lanes 0-15 / 16-31 both cover M=0..15; VGPR v holds
    // K = half*2 + v. Manual 1-deep pipeline on the A LDS loads so a
    // ds_load is always in flight across each WMMA pair; two accumulator
    // chains cover the WMMA->WMMA RAW latency.
    const float* arow = &wtile[t & 1][0] + col * E_DIM + half * 2;
    v8f c0 = {}, c1 = {};
    v2f a0 = *(const v2f*)(arow + 0);
    v2f a1 = *(const v2f*)(arow + 4);
#pragma unroll
    for (int kk = 0; kk < 32; kk += 2) {
      v2f n0 = a0, n1 = a1;
      if (kk + 2 < 32) {
        n0 = *(const v2f*)(arow + (kk + 2) * 4);
        n1 = *(const v2f*)(arow + (kk + 3) * 4);
      }
      c0 = __builtin_amdgcn_wmma_f32_16x16x4_f32(false, a0, false, bfrag[kk],
                                                 (short)0, c0, false, false);
      c1 = __builtin_amdgcn_wmma_f32_16x16x4_f32(false, a1, false, bfrag[kk + 1],
                                                 (short)0, c1, false, false);
      a0 = n0; a1 = n1;
    }

    // Per-lane running argmin over this tile's 8 codes.
    const int mBase = t * 16 + half * 8;
#pragma unroll
    for (int r = 0; r < 8; ++r) {
      float s = c0[r] + c1[r];
      float d = wnorm[mBase + r] - 2.0f * s;   // ||x||^2 dropped: same all k
      if (d < minv) { minv = d; mini = mBase + r; }  // ascending k order
    }
  }

  // Combine the two M-halves of the same x column (lane L <-> L^16).
  float ov = __shfl_xor(minv, 16, 32);
  int   oi = __shfl_xor(mini, 16, 32);
  if (ov < minv || (ov == minv && oi < mini)) { minv = ov; mini = oi; }
  if (half == 0) idx_out[rowBase + col] = mini;
}

// ---------------------------------------------------------------------------
// Kernel 3: z_q[n,:] = W[idx[n],:]   (one wave per row, W rows L2-hot)
// ---------------------------------------------------------------------------
__global__ __launch_bounds__(256) void vq_gather_kernel(
    const float* __restrict__ W, const int* __restrict__ idx,
    float* __restrict__ zq) {
  const int row  = blockIdx.x * 8 + (threadIdx.x >> 5);
  const int lane = threadIdx.x & 31;
  const int k    = idx[row];
  const float4* src = (const float4*)(W + (long)k * E_DIM);
  float4*       dst = (float4*)(zq + (long)row * E_DIM);
  dst[lane] = src[lane];
}

extern "C" void kernel_launch(void* const* d_in, const int* in_sizes, int n_in,
                              void* d_out, int out_size, void* d_ws, size_t ws_size,
                              hipStream_t stream) {
  const float* x = (const float*)d_in[0];   // [262144, 128] fp32
  const float* W = (const float*)d_in[1];   // [1024, 128]   fp32

  float* zq      = (float*)d_out;                               // z_q first
  int*   idx_out = (int*)((float*)d_out + (size_t)N_X * E_DIM); // then indices
  float* wnorm   = (float*)d_ws;                                // 1024 floats

  vq_wnorm_kernel <<<N_E / 4,  128, 0, stream>>>(W, wnorm);
  vq_argmin_kernel<<<N_X / 64, 128, 0, stream>>>(x, W, wnorm, idx_out);
  vq_gather_kernel<<<N_X / 8,  256, 0, stream>>>(W, idx_out, zq);
}